// DiffusioUnpool_17867063951389
// MI455X (gfx1250) — compile-verified
//
#include <hip/hip_runtime.h>

typedef __attribute__((ext_vector_type(2))) float v2f;
typedef __attribute__((ext_vector_type(8))) float v8f;

// ---------------------------------------------------------------- zero fill
__global__ void k_zero_f32(float* __restrict__ p, long long n) {
    long long i = (long long)blockIdx.x * blockDim.x + threadIdx.x;
    long long stride = (long long)gridDim.x * blockDim.x;
    for (; i < n; i += stride) p[i] = 0.0f;
}

__global__ void k_zero_f32x4(float4* __restrict__ p, long long n4) {
    long long i = (long long)blockIdx.x * blockDim.x + threadIdx.x;
    long long stride = (long long)gridDim.x * blockDim.x;
    float4 z = make_float4(0.f, 0.f, 0.f, 0.f);
    for (; i < n4; i += stride) p[i] = z;   // global_store_b128
}

// -------------------------------------------------- WMMA broadcast (step 1)
// win (n_win x node) = E(n_win x batch) @ nacmat(batch x node),
// E[r,b] = 1 iff r/S == b. v_wmma_f32_16x16x4_f32, one wave per 16x16 tile,
// K-loop over batch in steps of 4. f32 in/out -> exact vs the f32 reference.
__global__ __launch_bounds__(32) void k_win_nac_wmma(
        const float* __restrict__ nac, const int* __restrict__ d_batch,
        float* __restrict__ win, int n_perm, int nac_size) {
    const int batch  = d_batch[0];
    const int node   = nac_size / batch;
    const int n_win  = n_perm / node;
    const int S      = n_win / batch;
    const int tilesR = (n_win + 15) >> 4;
    const int tilesC = (node  + 15) >> 4;
    const int total  = tilesR * tilesC;

    const int lane = threadIdx.x;
    const int half = lane >> 4;   // 0: lanes 0-15, 1: lanes 16-31
    const int l15  = lane & 15;

    for (int t = blockIdx.x; t < total; t += gridDim.x) {
        const int r0 = (t / tilesC) * 16;
        const int n0 = (t % tilesC) * 16;
        v8f acc = {};
        for (int k0 = 0; k0 < batch; k0 += 4) {
            // A 16x4 f32: lane&15 = M; K = 2*half + vgpr
            const int r    = r0 + l15;
            const int bsel = (r < n_win) ? (r / S) : -1;
            const int ka   = k0 + 2 * half;
            v2f a;
            a.x = (bsel == ka     && ka     < batch) ? 1.0f : 0.0f;
            a.y = (bsel == ka + 1 && ka + 1 < batch) ? 1.0f : 0.0f;
            // B 4x16 f32: lane&15 = N; K = 2*half + vgpr (mirrors A layout)
            const int n  = n0 + l15;
            const int kb = k0 + 2 * half;
            v2f b;
            b.x = (kb     < batch && n < node) ? nac[kb * node + n]       : 0.0f;
            b.y = (kb + 1 < batch && n < node) ? nac[(kb + 1) * node + n] : 0.0f;
            acc = __builtin_amdgcn_wmma_f32_16x16x4_f32(
                false, a, false, b, (short)0, acc, false, false);
        }
        // C/D 16x16 f32: VGPR j -> M = j + 8*half, N = lane&15
        const int n = n0 + l15;
        if (n < node) {
#pragma unroll
            for (int j = 0; j < 8; ++j) {
                const int r = r0 + half * 8 + j;
                if (r < n_win) win[r * node + n] = acc[j];
            }
        }
    }
}

// ---------------------------------------------------------- degree of A
__global__ void k_rowsum_edges(const int* __restrict__ src,
                               const float* __restrict__ attr,
                               float* __restrict__ rowsum, int n_edges) {
    int e = blockIdx.x * blockDim.x + threadIdx.x;
    if (e < n_edges) atomicAdd(&rowsum[src[e]], attr[e]);
}

__global__ void k_dinv(const float* __restrict__ rowsum,
                       float* __restrict__ dinv, int N) {
    int i = blockIdx.x * blockDim.x + threadIdx.x;
    if (i < N) dinv[i] = rsqrtf(rowsum[i] + 1.0f);   // +1 = identity self-loop
}

// ================= generic (any F) scalar fallbacks =========================
__global__ void k_scatter_scaled(const float* __restrict__ fea,
                                 const int* __restrict__ perm,
                                 const float* __restrict__ win,
                                 const float* __restrict__ dinv,
                                 float* __restrict__ y,
                                 float* __restrict__ a_scaled,
                                 int n_perm, int F) {
    unsigned t = blockIdx.x * blockDim.x + threadIdx.x;
    unsigned total = (unsigned)n_perm * (unsigned)F;
    if (t >= total) return;
    int p = t / F, c = t % F;
    int r = perm[p];
    float dv = dinv[r];
    y[(long long)r * F + c] = fea[t] * dv;
    if (c == 0) a_scaled[r] = win[p] * dv;
}

__global__ void k_spmm_edges(const int* __restrict__ src,
                             const int* __restrict__ dst,
                             const float* __restrict__ attr,
                             const float* __restrict__ y,
                             const float* __restrict__ a_scaled,
                             float* __restrict__ accx,
                             float* __restrict__ acca,
                             int n_edges, int F) {
    unsigned t = blockIdx.x * blockDim.x + threadIdx.x;
    unsigned total = (unsigned)n_edges * (unsigned)F;
    if (t >= total) return;
    int e = t / F, c = t % F;
    int s = src[e], d = dst[e];
    float w = attr[e];
    atomicAdd(&accx[(long long)s * F + c], w * y[(long long)d * F + c]);
    if (c == 0) atomicAdd(&acca[s], w * a_scaled[d]);
}

__global__ void k_finalize(const float* __restrict__ dinv,
                           const float* __restrict__ y,
                           const float* __restrict__ a_scaled,
                           float* __restrict__ outx,
                           float* __restrict__ outa, int N, int F) {
    unsigned t = blockIdx.x * blockDim.x + threadIdx.x;
    unsigned total = (unsigned)N * (unsigned)F;
    if (t >= total) return;
    int i = t / F, c = t % F;
    float dv = dinv[i];
    outx[t] = dv * (outx[t] + y[t]);
    if (c == 0) outa[i] = dv * (outa[i] + a_scaled[i]);
}

// ================= F == 128 vectorized hot path (b128 VMEM, shift indexing) =
// 32 threads cover one row of 128 floats as float4 -> global_load/store_b128.
__global__ void k_scatter_scaled_f128(const float4* __restrict__ fea4,
                                      const int* __restrict__ perm,
                                      const float* __restrict__ win,
                                      const float* __restrict__ dinv,
                                      float4* __restrict__ y4,
                                      float* __restrict__ a_scaled,
                                      int n_perm) {
    unsigned t = blockIdx.x * blockDim.x + threadIdx.x;
    unsigned total = (unsigned)n_perm << 5;          // n_perm * 32
    if (t >= total) return;
    unsigned p = t >> 5, c4 = t & 31u;
    int r = perm[p];
    float dv = dinv[r];
    float4 v = fea4[t];
    v.x *= dv; v.y *= dv; v.z *= dv; v.w *= dv;
    y4[((unsigned long long)r << 5) + c4] = v;
    if (c4 == 0) a_scaled[r] = win[p] * dv;
}

__global__ void k_spmm_edges_f128(const int* __restrict__ src,
                                  const int* __restrict__ dst,
                                  const float* __restrict__ attr,
                                  const float4* __restrict__ y4,
                                  const float* __restrict__ a_scaled,
                                  float* __restrict__ accx,
                                  float* __restrict__ acca,
                                  int n_edges) {
    unsigned t = blockIdx.x * blockDim.x + threadIdx.x;
    unsigned total = (unsigned)n_edges << 5;         // n_edges * 32
    if (t >= total) return;
    unsigned e = t >> 5, c4 = t & 31u;
    int s = src[e], d = dst[e];
    float w = attr[e];
    float4 v = y4[((unsigned long long)d << 5) + c4];        // global_load_b128
    float* o = accx + ((unsigned long long)s << 7) + (c4 << 2);
    atomicAdd(o + 0, w * v.x);
    atomicAdd(o + 1, w * v.y);
    atomicAdd(o + 2, w * v.z);
    atomicAdd(o + 3, w * v.w);
    if (c4 == 0) atomicAdd(&acca[s], w * a_scaled[d]);
}

__global__ void k_finalize_f128(const float* __restrict__ dinv,
                                const float4* __restrict__ y4,
                                const float* __restrict__ a_scaled,
                                float4* __restrict__ outx4,
                                float* __restrict__ outa, int N) {
    unsigned t = blockIdx.x * blockDim.x + threadIdx.x;
    unsigned total = (unsigned)N << 5;               // N * 32
    if (t >= total) return;
    unsigned i = t >> 5, c4 = t & 31u;
    float dv = dinv[i];
    float4 a = outx4[t];
    float4 b = y4[t];
    a.x = dv * (a.x + b.x);
    a.y = dv * (a.y + b.y);
    a.z = dv * (a.z + b.z);
    a.w = dv * (a.w + b.w);
    outx4[t] = a;
    if (c4 == 0) outa[i] = dv * (outa[i] + a_scaled[i]);
}

extern "C" void kernel_launch(void* const* d_in, const int* in_sizes, int n_in,
                              void* d_out, int out_size, void* d_ws, size_t ws_size,
                              hipStream_t stream) {
    const float* fea   = (const float*)d_in[0];
    const int*   perm  = (const int*)d_in[1];
    const int*   eidx  = (const int*)d_in[2];
    const float* attr  = (const float*)d_in[3];
    const float* nac   = (const float*)d_in[4];
    const int*   d_bat = (const int*)d_in[6];
    (void)d_in[5]; (void)n_in; (void)ws_size;

    const int n_perm   = in_sizes[1];
    const int F        = in_sizes[0] / n_perm;   // fea_dim
    const int n_edges  = in_sizes[3];
    const int nac_size = in_sizes[4];
    const int N        = out_size / (F + 1);     // all_node_num

    const int* src = eidx;
    const int* dst = eidx + n_edges;

    // workspace (floats): rowsum[N] | dinv[N] | y[N*F] | a_scaled[N] | win[n_perm]
    float* ws       = (float*)d_ws;
    float* rowsum   = ws;
    float* dinv     = ws + N;
    float* y        = ws + 2LL * N;
    float* a_scaled = y + (long long)N * F;
    float* win      = a_scaled + N;

    float* outx = (float*)d_out;
    float* outa = outx + (long long)N * F;

    const int TB = 256;
    auto blocks = [](long long n, int tb) { return (unsigned)((n + tb - 1) / tb); };

    const bool vec = (F == 128) && (N % 4 == 0) && (n_perm % 4 == 0);

    // zero accumulators / scatter targets
    if (vec) {
        k_zero_f32x4<<<blocks(N / 4, TB), TB, 0, stream>>>((float4*)rowsum, N / 4);
        k_zero_f32x4<<<2048, TB, 0, stream>>>((float4*)y, ((long long)N * F + N) / 4);
        k_zero_f32x4<<<2048, TB, 0, stream>>>((float4*)outx, (long long)out_size / 4);
    } else {
        k_zero_f32<<<blocks(N, TB), TB, 0, stream>>>(rowsum, N);
        k_zero_f32<<<4096, TB, 0, stream>>>(y, (long long)N * F + N);
        k_zero_f32<<<4096, TB, 0, stream>>>(outx, (long long)out_size);
    }

    // step 1: WMMA broadcast ones@nac -> win
    k_win_nac_wmma<<<64, 32, 0, stream>>>(nac, d_bat, win, n_perm, nac_size);

    // degrees and D^{-1/2}
    k_rowsum_edges<<<blocks(n_edges, TB), TB, 0, stream>>>(src, attr, rowsum, n_edges);
    k_dinv<<<blocks(N, TB), TB, 0, stream>>>(rowsum, dinv, N);

    if (vec) {
        // unpool + right-scale (b128 loads/stores)
        k_scatter_scaled_f128<<<blocks((long long)n_perm * 32, TB), TB, 0, stream>>>(
            (const float4*)fea, perm, win, dinv, (float4*)y, a_scaled, n_perm);
        // sparse propagation (b128 gather of y[dst,:], scalar f32 atomics)
        k_spmm_edges_f128<<<blocks((long long)n_edges * 32, TB), TB, 0, stream>>>(
            src, dst, attr, (const float4*)y, a_scaled, outx, outa, n_edges);
        // self-loop + left-scale (b128)
        k_finalize_f128<<<blocks((long long)N * 32, TB), TB, 0, stream>>>(
            dinv, (const float4*)y, a_scaled, (float4*)outx, outa, N);
    } else {
        k_scatter_scaled<<<blocks((long long)n_perm * F, TB), TB, 0, stream>>>(
            fea, perm, win, dinv, y, a_scaled, n_perm, F);
        k_spmm_edges<<<blocks((long long)n_edges * F, TB), TB, 0, stream>>>(
            src, dst, attr, y, a_scaled, outx, outa, n_edges, F);
        k_finalize<<<blocks((long long)N * F, TB), TB, 0, stream>>>(
            dinv, y, a_scaled, outx, outa, N, F);
    }
}